// GMKANCEll_44315472560858
// MI455X (gfx1250) — compile-verified
//
#include <hip/hip_runtime.h>
#include <hip/hip_bf16.h>

// ---------------------------------------------------------------------------
// GMKANCEll forward for MI455X (gfx1250, wave32, WMMA).
//
// Heavy part: three KAN experts = one fused GEMM
//   out[b,o] = sum_i sum_f Basis[b,i,f] * W[i,f,o],  f in [0,16)
//   f 0..3  : Chebyshev T_k(tanh(xc))      (weights cheb_c[i,o,k], k contig)
//   f 4..11 : cubic B-spline bases of xb   (weights spline_w[o,i,k], k contig)
//   f 12..14: xt, xt^2, xt^3               (weights taylor_c[i,o,p], p contig)
//   f 15    : silu(xb)                     (weights base_w[o,i])
// 656 MB of f32 weights streamed exactly once (HBM floor ~28us @ 23.3 TB/s);
// basis (6.5 MB) stays in the 192 MB L2 across the 200 o-tiles.
// Matrix op: V_WMMA_F32_16X16X4_F32, K=4 per instruction, f32 accumulate.
// Each wave owns one K-chunk and BOTH M-tiles (two accumulators), so every
// weight fragment is loaded once and feeds two WMMAs -> weight-load
// instruction stream halved vs. one-M-tile-per-wave, and two independent
// accumulation chains hide f32-WMMA accumulation latency.
// ---------------------------------------------------------------------------

typedef float v2f __attribute__((ext_vector_type(2)));
typedef float v8f __attribute__((ext_vector_type(8)));

#define Bn   32
#define Nn   100
#define Hn   32
#define Dn   3200
#define NW   1596      // (3200-10)/2 + 1
#define KSPL 8         // K-chunks (one per wave) in expert kernel
#define ICH  (Dn / KSPL)

// ---------------- kernel 0: alpha[n] = sigmoid(mean_b inputs[b,n]) ----------
__global__ void k_alpha(const float* __restrict__ inputs, float* __restrict__ alpha) {
  int n = threadIdx.x;
  if (n < Nn) {
    float s = 0.f;
    for (int b = 0; b < Bn; ++b) s += inputs[b * Nn + n];
    s *= (1.f / Bn);
    alpha[n] = 1.f / (1.f + __expf(-s));
  }
}

// ---------------- kernel 1: graph conv + layernorm -> hs[b, m*32+o] ---------
__global__ __launch_bounds__(32) void k_gconv(
    const float* __restrict__ inputs, const float* __restrict__ hidden,
    const float* __restrict__ lap, const float* __restrict__ corr,
    const float* __restrict__ gc_w, const float* __restrict__ gc_b,
    const float* __restrict__ ln_g, const float* __restrict__ ln_b,
    const float* __restrict__ alpha, float* __restrict__ hs) {
  __shared__ float sax[33];
  const int bm = blockIdx.x;
  const int b = bm / Nn, m = bm % Nn;
  const int lane = threadIdx.x;

  // ax[f] = sum_n (alpha[n]*lap[m,n] + (1-alpha[n])*corr[m,n]) * concat[b,n,f]
  for (int f = lane; f < 33; f += 32) {
    float s = 0.f;
    for (int nn = 0; nn < Nn; ++nn) {
      float an = alpha[nn];
      float A  = an * lap[m * Nn + nn] + (1.f - an) * corr[m * Nn + nn];
      float cv = (f == 0) ? inputs[b * Nn + nn] : hidden[(b * Nn + nn) * Hn + (f - 1)];
      s = fmaf(A, cv, s);
    }
    sax[f] = s;
  }
  __syncthreads();

  float h = gc_b[lane];
  for (int f = 0; f < 33; ++f) h = fmaf(sax[f], gc_w[f * Hn + lane], h);

  // layernorm over the 32 lanes (wave32 shuffle reduction)
  float mu = h;
  for (int msk = 16; msk >= 1; msk >>= 1) mu += __shfl_xor(mu, msk, 32);
  mu *= (1.f / 32.f);
  float dv = (h - mu) * (h - mu);
  for (int msk = 16; msk >= 1; msk >>= 1) dv += __shfl_xor(dv, msk, 32);
  dv *= (1.f / 32.f);
  float y = (h - mu) * rsqrtf(dv + 1e-5f) * ln_g[lane] + ln_b[lane];
  hs[(long long)b * Dn + m * Hn + lane] = y;
}

// ---------------- kernel 2: gating scores -> gates[b,3] ---------------------
__global__ __launch_bounds__(256) void k_gates(
    const float* __restrict__ hs, const float* __restrict__ min_p,
    const float* __restrict__ max_p, const float* __restrict__ min_s,
    const float* __restrict__ max_s, const float* __restrict__ min_n,
    const float* __restrict__ max_n, float* __restrict__ gates) {
  __shared__ float red[3 * 256];
  const int b = blockIdx.x, tid = threadIdx.x;
  const float* x = hs + (long long)b * Dn;

  float c10[10], s10[10];
  for (int t = 0; t < 10; ++t) {
    float ang = -6.2831853071795864f * (float)t * 0.1f;
    __sincosf(ang, &s10[t], &c10[t]);
  }

  float psum = 0.f, ssum = 0.f;
  for (int w = tid; w < NW; w += 256) {
    float win[10];
    for (int j = 0; j < 10; ++j) win[j] = x[w * 2 + j];
    for (int k = 0; k < 10; ++k) {
      float re = 0.f, im = 0.f;
      int t = 0;
      for (int j = 0; j < 10; ++j) {
        re = fmaf(win[j], c10[t], re);
        im = fmaf(win[j], s10[t], im);
        t += k; if (t >= 10) t -= 10;
      }
      psum += sqrtf(re * re + im * im);
    }
    for (int j = 0; j < 9; ++j) ssum += fabsf(win[j + 1] - win[j]);
  }
  float nsum = 0.f;
  for (int i = tid; i < Dn - 2; i += 256)
    nsum += fabsf(x[i + 2] - 2.f * x[i + 1] + x[i]);

  red[tid] = psum; red[256 + tid] = ssum; red[512 + tid] = nsum;
  __syncthreads();
  for (int st = 128; st >= 1; st >>= 1) {
    if (tid < st) {
      red[tid]       += red[tid + st];
      red[256 + tid] += red[256 + tid + st];
      red[512 + tid] += red[512 + tid + st];
    }
    __syncthreads();
  }
  if (tid == 0) {
    float pm = red[0]   / ((float)NW * 10.f);
    float sm = red[256] / ((float)NW * 9.f);
    float nm = red[512] / (float)(Dn - 2);
    float p  = fabsf((pm - min_p[0]) / (max_p[0] - min_p[0] + 1e-8f));
    float s  = 1.f - fabsf((sm - min_s[0]) / (max_s[0] - min_s[0] + 1e-8f));
    float nv = fabsf((nm - min_n[0]) / (max_n[0] - min_n[0] + 1e-8f));
    float a0 = fabsf(p), a1 = fabsf(s), a2 = fabsf(nv);
    float mx = fmaxf(a0, fmaxf(a1, a2));
    float e0 = __expf(a0 - mx), e1 = __expf(a1 - mx), e2 = __expf(a2 - mx);
    float inv = 1.f / (e0 + e1 + e2);
    gates[b * 3 + 0] = e0 * inv;
    gates[b * 3 + 1] = e1 * inv;
    gates[b * 3 + 2] = e2 * inv;
  }
}

// ---------------- kernel 3: 16 basis features per (b,i) ---------------------
__global__ __launch_bounds__(256) void k_basis(
    const float* __restrict__ hs, const float* __restrict__ gates,
    float* __restrict__ basis) {
  int idx = blockIdx.x * 256 + threadIdx.x;
  if (idx >= Bn * Dn) return;
  int b = idx / Dn;
  float x  = hs[idx];
  float xc = gates[b * 3 + 0] * x;
  float xb = gates[b * 3 + 1] * x;
  float xt = gates[b * 3 + 2] * x;

  float f[16];
  // Chebyshev: cos(k*acos(tanh xc)) == T_k(tanh xc)  (recurrence, no acos)
  float t = tanhf(xc);
  f[0] = 1.f; f[1] = t; f[2] = 2.f * t * t - 1.f; f[3] = t * (4.f * t * t - 3.f);

  // cubic B-spline bases on uniform grid g_j = -2.2 + 0.4*j, j=0..11
  float bb[11];
  for (int j = 0; j < 11; ++j) {
    float gl = fmaf((float)j, 0.4f, -2.2f);
    bb[j] = (xb >= gl && xb < gl + 0.4f) ? 1.f : 0.f;
  }
  for (int d = 1; d <= 3; ++d) {
    float invd = 1.f / (0.4f * (float)d);
    for (int j = 0; j < 11 - d; ++j) {
      float gj  = fmaf((float)j, 0.4f, -2.2f);
      float gjd = fmaf((float)(j + d + 1), 0.4f, -2.2f);
      bb[j] = (xb - gj) * invd * bb[j] + (gjd - xb) * invd * bb[j + 1];
    }
  }
  for (int k = 0; k < 8; ++k) f[4 + k] = bb[k];

  f[12] = xt; f[13] = xt * xt; f[14] = xt * xt * xt;
  f[15] = xb / (1.f + __expf(-xb));   // silu

  float4* dst = (float4*)(basis + (long long)idx * 16);
  dst[0] = make_float4(f[0], f[1], f[2], f[3]);
  dst[1] = make_float4(f[4], f[5], f[6], f[7]);
  dst[2] = make_float4(f[8], f[9], f[10], f[11]);
  dst[3] = make_float4(f[12], f[13], f[14], f[15]);
}

// ---------------- kernel 4: fused expert GEMM via V_WMMA_F32_16X16X4_F32 ----
// grid.x = 200 o-tiles of 16. 8 waves/block, wave = K-chunk (400 i each).
// Each wave accumulates BOTH M-tiles (batch 0..15 and 16..31): weight (B)
// fragments are loaded once per i and feed two WMMAs. Per i: 8 WMMA
// (cheb, spline-lo, spline-hi, taylor+silu*base) x 2 M-tiles, f32 accum.
// K-chunk partials merged with ds_add_f32 LDS atomics, then coalesced store.
__global__ __launch_bounds__(256) void k_expert(
    const float* __restrict__ basis, const float* __restrict__ cheb,
    const float* __restrict__ spline, const float* __restrict__ taylor,
    const float* __restrict__ basew, float* __restrict__ out) {
  __shared__ float acc[Bn * 16];
  const int tid  = threadIdx.x;
  const int wave = tid >> 5, lane = tid & 31;
  const int kc = wave;                 // 0..7 -> K-chunk
  const int n = lane & 15, half = lane >> 4;
  const int o0 = blockIdx.x * 16, o = o0 + n;

  for (int tix = tid; tix < Bn * 16; tix += 256) acc[tix] = 0.f;
  __syncthreads();

  const long long D  = Dn;
  const int       i0 = kc * ICH;
  // A: 16x4 f32 layout -> lane%16 = M row, lane/16 selects K pair {0,1}/{2,3}
  const float* pA0 = basis + (((long long)n)        * D + i0) * 16 + half * 2; // batch 0..15
  const float* pA1 = basis + (((long long)(16 + n)) * D + i0) * 16 + half * 2; // batch 16..31
  // B: 4x16 f32 layout -> lane%16 = N col, lane/16 selects K pair
  const float* pC = cheb   + (((long long)i0) * D + o) * 4 + half * 2; // cheb_c[i,o,k]
  const float* pS = spline + (((long long)o) * D + i0) * 8 + half * 2; // spline_w[o,i,k]
  const float* pT = taylor + (((long long)i0) * D + o) * 3;            // taylor_c[i,o,p]
  const float* pB = basew  + ((long long)o) * D + i0;                  // base_w[o,i]

  v8f c0 = {0.f, 0.f, 0.f, 0.f, 0.f, 0.f, 0.f, 0.f};
  v8f c1 = {0.f, 0.f, 0.f, 0.f, 0.f, 0.f, 0.f, 0.f};

  for (int i = 0; i < ICH; ++i) {
    // A fragments, both M-tiles (basis is L2-resident -> cheap)
    v2f a0_ch = *(const v2f*)(pA0 + 0);
    v2f a0_s0 = *(const v2f*)(pA0 + 4);
    v2f a0_s1 = *(const v2f*)(pA0 + 8);
    v2f a0_tb = *(const v2f*)(pA0 + 12);
    v2f a1_ch = *(const v2f*)(pA1 + 0);
    v2f a1_s0 = *(const v2f*)(pA1 + 4);
    v2f a1_s1 = *(const v2f*)(pA1 + 8);
    v2f a1_tb = *(const v2f*)(pA1 + 12);

    // B (weight) fragments, loaded ONCE, used by both M-tiles
    v2f b_ch = *(const v2f*)(pC);
    v2f b_s0 = *(const v2f*)(pS);
    v2f b_s1 = *(const v2f*)(pS + 4);
    v2f b_tb;
    b_tb.x = pT[half * 2];              // taylor p0 / p2
    b_tb.y = half ? pB[0] : pT[1];      // base_w  / taylor p1

    c0 = __builtin_amdgcn_wmma_f32_16x16x4_f32(false, a0_ch, false, b_ch, (short)0, c0, false, false);
    c1 = __builtin_amdgcn_wmma_f32_16x16x4_f32(false, a1_ch, false, b_ch, (short)0, c1, false, false);
    c0 = __builtin_amdgcn_wmma_f32_16x16x4_f32(false, a0_s0, false, b_s0, (short)0, c0, false, false);
    c1 = __builtin_amdgcn_wmma_f32_16x16x4_f32(false, a1_s0, false, b_s0, (short)0, c1, false, false);
    c0 = __builtin_amdgcn_wmma_f32_16x16x4_f32(false, a0_s1, false, b_s1, (short)0, c0, false, false);
    c1 = __builtin_amdgcn_wmma_f32_16x16x4_f32(false, a1_s1, false, b_s1, (short)0, c1, false, false);
    c0 = __builtin_amdgcn_wmma_f32_16x16x4_f32(false, a0_tb, false, b_tb, (short)0, c0, false, false);
    c1 = __builtin_amdgcn_wmma_f32_16x16x4_f32(false, a1_tb, false, b_tb, (short)0, c1, false, false);

    pA0 += 16; pA1 += 16; pC += 4 * D; pS += 8; pT += 3 * D; pB += 1;
  }

  // C/D layout: lanes 0-15 -> M = v, lanes 16-31 -> M = v+8, N = lane%16
  for (int v = 0; v < 8; ++v) {
    int m = half * 8 + v;
    atomicAdd(&acc[m * 16 + n], c0[v]);          // ds_add_f32, batch 0..15
    atomicAdd(&acc[(16 + m) * 16 + n], c1[v]);   // ds_add_f32, batch 16..31
  }
  __syncthreads();

  for (int tix = tid; tix < Bn * 16; tix += 256) {
    int brow = tix >> 4, col = tix & 15;
    out[(long long)brow * D + o0 + col] = acc[tix];
  }
}

// ---------------------------------------------------------------------------
extern "C" void kernel_launch(void* const* d_in, const int* in_sizes, int n_in,
                              void* d_out, int out_size, void* d_ws, size_t ws_size,
                              hipStream_t stream) {
  const float* inputs   = (const float*)d_in[0];
  const float* hidden   = (const float*)d_in[1];
  const float* lap      = (const float*)d_in[2];
  const float* corr     = (const float*)d_in[3];
  const float* gc_w     = (const float*)d_in[4];
  const float* gc_b     = (const float*)d_in[5];
  const float* ln_g     = (const float*)d_in[6];
  const float* ln_b     = (const float*)d_in[7];
  const float* cheb_c   = (const float*)d_in[8];
  const float* base_w   = (const float*)d_in[9];
  const float* spline_w = (const float*)d_in[10];
  const float* taylor_c = (const float*)d_in[11];
  const float* min_p    = (const float*)d_in[12];
  const float* max_p    = (const float*)d_in[13];
  const float* min_s    = (const float*)d_in[14];
  const float* max_s    = (const float*)d_in[15];
  const float* min_n    = (const float*)d_in[16];
  const float* max_n    = (const float*)d_in[17];

  float* ws    = (float*)d_ws;
  float* alpha = ws;                         // 100 (pad 128)
  float* hs    = ws + 128;                   // 32*3200
  float* gates = hs + Bn * Dn;               // 96 (pad 128)
  float* basis = gates + 128;                // 32*3200*16 (16B aligned)

  k_alpha<<<1, 128, 0, stream>>>(inputs, alpha);
  k_gconv<<<Bn * Nn, 32, 0, stream>>>(inputs, hidden, lap, corr, gc_w, gc_b,
                                      ln_g, ln_b, alpha, hs);
  k_gates<<<Bn, 256, 0, stream>>>(hs, min_p, max_p, min_s, max_s, min_n, max_n, gates);
  k_basis<<<(Bn * Dn + 255) / 256, 256, 0, stream>>>(hs, gates, basis);
  k_expert<<<Dn / 16, 256, 0, stream>>>(basis, cheb_c, spline_w, taylor_c, base_w,
                                        (float*)d_out);
}